// DeltaNetAttention_42666205118980
// MI455X (gfx1250) — compile-verified
//
#include <hip/hip_runtime.h>
#include <hip/hip_bf16.h>
#include <math.h>

// Problem constants (from reference)
#define B_    256
#define HID_  2048
#define QKV_  6144
#define GATE_ 2048
#define NH_   16
#define HD_   128

typedef __attribute__((ext_vector_type(2))) float v2f;
typedef __attribute__((ext_vector_type(8))) float v8f;

__device__ __forceinline__ float sigmoid_(float x) { return 1.0f / (1.0f + expf(-x)); }
__device__ __forceinline__ float silu_(float x)    { return x * sigmoid_(x); }
__device__ __forceinline__ float softplus_(float x){ return (x > 30.0f) ? x : log1pf(expf(x)); }

// ---------------------------------------------------------------------------
// Generic C[m,n] = sum_k X[m,k] * W[n,k]   (X: MxK row-major, W: NxK row-major)
// One wave computes one 16x16 output tile with V_WMMA_F32_16X16X4_F32.
// A (16x4 f32): lanes 0-15 hold M=lane, lanes 16-31 hold M=lane-16;
//               VGPR0 = K=0 / K=2, VGPR1 = K=1 / K=3  -> one float2 load/lane.
// B (4x16 f32): symmetric with N on lanes.
// D (16x16 f32, 8 VGPRs): VGPR r -> M = r + (lane<16?0:8), N = lane%16.
// ---------------------------------------------------------------------------
__global__ __launch_bounds__(256) void wmma_gemm_xwt(
    const float* __restrict__ X, const float* __restrict__ W,
    float* __restrict__ C, int M, int N, int K)
{
    const int wave = threadIdx.x >> 5;
    const int lane = threadIdx.x & 31;
    const int m0 = blockIdx.y * 16;
    const int n0 = (blockIdx.x * 8 + wave) * 16;
    if (n0 >= N || m0 >= M) return;   // wave-uniform exit: EXEC stays all-1s

    const int lr = lane & 15;          // row (A:M, B:N) within tile
    const int kh = (lane >> 4) * 2;    // K sub-offset 0 or 2
    const float* a_ptr = X + (size_t)(m0 + lr) * K + kh;
    const float* b_ptr = W + (size_t)(n0 + lr) * K + kh;

    v8f acc = {0.f, 0.f, 0.f, 0.f, 0.f, 0.f, 0.f, 0.f};
#pragma unroll 4
    for (int k = 0; k < K; k += 4) {
        v2f a = *(const v2f*)(a_ptr + k);
        v2f b = *(const v2f*)(b_ptr + k);
        acc = __builtin_amdgcn_wmma_f32_16x16x4_f32(
            /*neg_a=*/false, a, /*neg_b=*/false, b,
            /*c_mod=*/(short)0, acc, /*reuse_a=*/false, /*reuse_b=*/false);
    }

    const int cn = n0 + lr;
    const int cm = m0 + (lane >> 4) * 8;
#pragma unroll
    for (int r = 0; r < 8; ++r)
        C[(size_t)(cm + r) * N + cn] = acc[r];
}

// ---------------------------------------------------------------------------
// Depthwise conv (K=4) + SiLU; also emits new_conv_state = [cs1, cs2, qkv].
// ---------------------------------------------------------------------------
__global__ __launch_bounds__(256) void conv_silu_kernel(
    const float* __restrict__ conv_state,  // B x QKV x 3
    const float* __restrict__ qkv,         // B x QKV
    const float* __restrict__ conv_w,      // QKV x 4
    float* __restrict__ qkvconv,           // B x QKV (silu'd)
    float* __restrict__ new_conv_state)    // B x QKV x 3 (-> d_out)
{
    const size_t idx = (size_t)blockIdx.x * blockDim.x + threadIdx.x;
    if (idx >= (size_t)B_ * QKV_) return;
    const int c = (int)(idx % QKV_);

    const float* cs = conv_state + idx * 3;
    const float s0 = cs[0], s1 = cs[1], s2 = cs[2];
    const float x3 = qkv[idx];
    const float* w = conv_w + (size_t)c * 4;
    const float acc = s0 * w[0] + s1 * w[1] + s2 * w[2] + x3 * w[3];
    qkvconv[idx] = silu_(acc);

    float* ncs = new_conv_state + idx * 3;
    ncs[0] = s1; ncs[1] = s2; ncs[2] = x3;
}

// ---------------------------------------------------------------------------
// Per-(b,h) delta-rule state update. One 128-thread workgroup per (b,h).
// Thread j owns column j of the 128x128 state.
// ---------------------------------------------------------------------------
__global__ __launch_bounds__(128) void state_update_kernel(
    const float* __restrict__ qkvconv,   // B x 6144 (q|k|v)
    const float* __restrict__ Sin,       // B x NH x HD x HD
    const float* __restrict__ aproj,     // B x NH
    const float* __restrict__ bproj,     // B x NH
    const float* __restrict__ dt_bias,   // NH
    const float* __restrict__ A_log,     // NH
    const float* __restrict__ norm_w,    // HD
    const float* __restrict__ zraw,      // B x 2048 (pre-silu)
    float* __restrict__ Sout,            // -> d_out S region
    float* __restrict__ og)              // B x 2048 gated output for final GEMM
{
    __shared__ float s_kn[HD_];
    __shared__ float s_qn[HD_];
    __shared__ float s_red[HD_];

    const int b = blockIdx.x >> 4;
    const int h = blockIdx.x & 15;
    const int j = threadIdx.x;

    const float* qb = qkvconv + (size_t)b * QKV_ + h * HD_;
    const float qv = qb[j];
    const float kv = qb[2048 + j];
    const float vv = qb[4096 + j];

    // block reductions for ||q||, ||k||
    s_red[j] = qv * qv; __syncthreads();
    for (int s = 64; s > 0; s >>= 1) { if (j < s) s_red[j] += s_red[j + s]; __syncthreads(); }
    const float qss = s_red[0]; __syncthreads();
    s_red[j] = kv * kv; __syncthreads();
    for (int s = 64; s > 0; s >>= 1) { if (j < s) s_red[j] += s_red[j + s]; __syncthreads(); }
    const float kss = s_red[0]; __syncthreads();

    const float qscale = 1.0f / sqrtf((float)HD_);
    s_qn[j] = qv / fmaxf(sqrtf(qss), 1e-12f) * qscale;
    s_kn[j] = kv / fmaxf(sqrtf(kss), 1e-12f);
    __syncthreads();

    const float beta = sigmoid_(bproj[b * NH_ + h]);
    const float aval = aproj[b * NH_ + h] + dt_bias[h];
    const float g    = -expf(A_log[h]) * softplus_(aval);
    const float gexp = expf(g);

    const float* Sb = Sin  + ((size_t)(b * NH_ + h)) * HD_ * HD_;
    float*       So = Sout + ((size_t)(b * NH_ + h)) * HD_ * HD_;

    // pass 1: kv_mem[j] = gexp * sum_i S[i][j] * kn[i]
    float kvmem = 0.0f;
#pragma unroll 4
    for (int i = 0; i < HD_; ++i)
        kvmem += Sb[(size_t)i * HD_ + j] * s_kn[i];
    kvmem *= gexp;
    const float delta = (vv - kvmem) * beta;

    // pass 2: S_new[i][j] = S[i][j]*gexp + kn[i]*delta ; o[j] = sum_i qn[i]*S_new[i][j]
    float o = 0.0f;
#pragma unroll 4
    for (int i = 0; i < HD_; ++i) {
        const float s = Sb[(size_t)i * HD_ + j] * gexp + s_kn[i] * delta;
        So[(size_t)i * HD_ + j] = s;
        o += s_qn[i] * s;
    }

    // RMS norm over j + gate
    s_red[j] = o * o; __syncthreads();
    for (int s = 64; s > 0; s >>= 1) { if (j < s) s_red[j] += s_red[j + s]; __syncthreads(); }
    const float var = s_red[0] * (1.0f / HD_);
    float on = o * rsqrtf(var + 1e-6f) * norm_w[j];

    const float zr = zraw[(size_t)b * GATE_ + h * HD_ + j];
    const float z1 = silu_(zr);       // z = silu(x0 @ W_z.T)
    on *= silu_(z1);                  // o_n * silu(z)  (double silu per reference)
    og[(size_t)b * GATE_ + h * HD_ + j] = on;
}

// ---------------------------------------------------------------------------
extern "C" void kernel_launch(void* const* d_in, const int* in_sizes, int n_in,
                              void* d_out, int out_size, void* d_ws, size_t ws_size,
                              hipStream_t stream) {
    const float* x        = (const float*)d_in[0];   // B x 1 x HID
    const float* cstate   = (const float*)d_in[1];   // B x QKV x 3
    const float* rstate   = (const float*)d_in[2];   // B x NH x HD x HD
    const float* W_qkv    = (const float*)d_in[3];   // QKV x HID
    const float* W_z      = (const float*)d_in[4];   // GATE x HID
    const float* W_b      = (const float*)d_in[5];   // NH x HID
    const float* W_a      = (const float*)d_in[6];   // NH x HID
    const float* conv_w   = (const float*)d_in[7];   // QKV x 1 x 4
    const float* dt_bias  = (const float*)d_in[8];   // NH
    const float* A_log    = (const float*)d_in[9];   // NH
    const float* norm_w   = (const float*)d_in[10];  // HD
    const float* W_out    = (const float*)d_in[11];  // HID x GATE

    float* out = (float*)d_out;                                   // B x HID
    float* new_conv_state = out + (size_t)B_ * HID_;              // B x QKV x 3
    float* Sout = new_conv_state + (size_t)B_ * QKV_ * 3;         // B x NH x HD x HD

    // workspace layout (floats)
    float* ws       = (float*)d_ws;
    float* qkv      = ws;                                         // B x QKV
    float* zraw     = qkv + (size_t)B_ * QKV_;                    // B x GATE
    float* bproj    = zraw + (size_t)B_ * GATE_;                  // B x NH
    float* aproj    = bproj + (size_t)B_ * NH_;                   // B x NH
    float* qkvconv  = aproj + (size_t)B_ * NH_;                   // B x QKV
    float* og       = qkvconv + (size_t)B_ * QKV_;                // B x GATE

    dim3 blk(256);

    // 1) input projections (fp32 WMMA GEMMs)
    {
        dim3 g((QKV_ / 16 + 7) / 8, B_ / 16);
        wmma_gemm_xwt<<<g, blk, 0, stream>>>(x, W_qkv, qkv, B_, QKV_, HID_);
    }
    {
        dim3 g((GATE_ / 16 + 7) / 8, B_ / 16);
        wmma_gemm_xwt<<<g, blk, 0, stream>>>(x, W_z, zraw, B_, GATE_, HID_);
    }
    {
        dim3 g(1, B_ / 16);
        wmma_gemm_xwt<<<g, blk, 0, stream>>>(x, W_b, bproj, B_, NH_, HID_);
        wmma_gemm_xwt<<<g, blk, 0, stream>>>(x, W_a, aproj, B_, NH_, HID_);
    }

    // 2) depthwise conv + silu (+ new_conv_state)
    {
        size_t n = (size_t)B_ * QKV_;
        conv_silu_kernel<<<dim3((n + 255) / 256), blk, 0, stream>>>(
            cstate, qkv, conv_w, qkvconv, new_conv_state);
    }

    // 3) per-(b,h) state update + RMS norm + gate
    state_update_kernel<<<dim3(B_ * NH_), dim3(128), 0, stream>>>(
        qkvconv, rstate, aproj, bproj, dt_bias, A_log, norm_w, zraw, Sout, og);

    // 4) output projection
    {
        dim3 g((HID_ / 16 + 7) / 8, B_ / 16);
        wmma_gemm_xwt<<<g, blk, 0, stream>>>(og, W_out, out, B_, HID_, GATE_);
    }

    (void)in_sizes; (void)n_in; (void)out_size; (void)ws_size;
}